// Quantize_61177514164441
// MI455X (gfx1250) — compile-verified
//
#include <hip/hip_runtime.h>

#define B_ROWS 32768
#define K_CODES 2048
#define D_DIM 32
#define LOSS_IDX (B_ROWS * D_DIM + B_ROWS)   // emb_out then sk_ids then loss

constexpr double SK_EPS_D = 0.003;
constexpr int SK_ITERS_N = 100;

typedef float v2f __attribute__((ext_vector_type(2)));
typedef float v8f __attribute__((ext_vector_type(8)));

// ---- monotonic float <-> uint mapping for atomic min/max ----
__device__ __forceinline__ unsigned fflip(float f) {
    unsigned u = __float_as_uint(f);
    return u ^ ((u >> 31) ? 0xFFFFFFFFu : 0x80000000u);
}
__device__ __forceinline__ float funflip(unsigned u) {
    return __uint_as_float(u ^ ((u >> 31) ? 0x80000000u : 0xFFFFFFFFu));
}

// ---- init: v=1, vacc=0, min/max sentinels, loss slot = 0 ----
__global__ void init_kernel(double* __restrict__ v, double* __restrict__ vacc,
                            unsigned* __restrict__ mm, float* __restrict__ out) {
    int j = blockIdx.x * 256 + threadIdx.x;
    if (j < K_CODES) { v[j] = 1.0; vacc[j] = 0.0; }
    if (j == 0) {
        mm[0] = 0xFFFFFFFFu;  // flipped +inf (min accumulator)
        mm[1] = 0u;           // flipped -inf (max accumulator)
        out[LOSS_IDX] = 0.0f;
    }
}

// ---- row norms ||x_i||^2 and ||c_k||^2 ----
__global__ void norms_kernel(const float* __restrict__ x, const float* __restrict__ w,
                             float* __restrict__ xn, float* __restrict__ cn) {
    int i = blockIdx.x * 256 + threadIdx.x;
    if (i < B_ROWS) {
        float s = 0.f;
        #pragma unroll
        for (int d = 0; d < D_DIM; d++) { float t = x[(size_t)i * D_DIM + d]; s += t * t; }
        xn[i] = s;
    }
    if (i < K_CODES) {
        float s = 0.f;
        #pragma unroll
        for (int d = 0; d < D_DIM; d++) { float t = w[(size_t)i * D_DIM + d]; s += t * t; }
        cn[i] = s;
    }
}

// ---- dist = ||x||^2 + ||c||^2 - 2 x.c  via V_WMMA_F32_16X16X4_F32 ----
// One wave per 16x16 tile; 8 WMMAs over K=32.
__global__ __launch_bounds__(256) void dist_wmma_kernel(
    const float* __restrict__ x, const float* __restrict__ w,
    const float* __restrict__ xn, const float* __restrict__ cn,
    float* __restrict__ dist, unsigned* __restrict__ mm) {
    int wid  = blockIdx.x * 8 + (threadIdx.x >> 5);
    int lane = threadIdx.x & 31;
    int tM = wid >> 7;          // K_CODES/16 = 128 tiles in N
    int tN = wid & 127;
    int row0 = tM * 16, col0 = tN * 16;

    // A 16x4 f32 layout: lanes 0-15 hold M=0..15 (K=0,1 in v0,v1); lanes 16-31 K=2,3
    int m  = lane & 15;
    int kb = (lane >> 4) << 1;  // 0 or 2
    const float* xa = x + (size_t)(row0 + m) * D_DIM + kb;
    // B 4x16: B[d][n] = w[n][d]; lane holds n = lane&15, K pair per half-wave
    const float* wb = w + (size_t)(col0 + m) * D_DIM + kb;

    v8f acc = {};
    #pragma unroll
    for (int d0 = 0; d0 < D_DIM; d0 += 4) {
        v2f a = *(const v2f*)(xa + d0);
        v2f b = *(const v2f*)(wb + d0);
        acc = __builtin_amdgcn_wmma_f32_16x16x4_f32(false, a, false, b,
                                                    (short)0, acc, false, false);
    }

    // C/D layout: VGPR r -> M=r (lanes 0-15) / M=r+8 (lanes 16-31), N = lane&15
    int mOff = (lane >> 4) << 3;
    int n = lane & 15;
    float lmin = 3.4e38f, lmax = -3.4e38f;
    #pragma unroll
    for (int r = 0; r < 8; r++) {
        int row = row0 + r + mOff;
        int col = col0 + n;
        float dv = xn[row] + cn[col] - 2.0f * acc[r];
        dist[(size_t)row * K_CODES + col] = dv;
        lmin = fminf(lmin, dv);
        lmax = fmaxf(lmax, dv);
    }
    #pragma unroll
    for (int off = 16; off; off >>= 1) {
        lmin = fminf(lmin, __shfl_xor(lmin, off, 32));
        lmax = fmaxf(lmax, __shfl_xor(lmax, off, 32));
    }
    if (lane == 0) {
        atomicMin(&mm[0], fflip(lmin));
        atomicMax(&mm[1], fflip(lmax));
    }
}

// ---- Km = exp(-((dist-mid)/amp)/eps) in f64 (range e^±333 needs f64) ----
__global__ __launch_bounds__(256) void km_kernel(const float* __restrict__ dist,
                                                 const unsigned* __restrict__ mm,
                                                 double* __restrict__ Km) {
    size_t q = (size_t)blockIdx.x * 256 + threadIdx.x;   // 4 elements per thread
    float mn = funflip(mm[0]);
    float mx = funflip(mm[1]);
    float mid = (mx + mn) * 0.5f;
    float amp = mx - mid + 1e-5f;
    double scale = -1.0 / ((double)amp * SK_EPS_D);
    double bias  = -(double)mid * scale;
    float4 dv = ((const float4*)dist)[q];
    size_t base = q * 4;
    Km[base + 0] = exp((double)dv.x * scale + bias);
    Km[base + 1] = exp((double)dv.y * scale + bias);
    Km[base + 2] = exp((double)dv.z * scale + bias);
    Km[base + 3] = exp((double)dv.w * scale + bias);
}

// ---- fused Sinkhorn pass: one read of Km gives both u and K^T u ----
// b128 loads, software-pipelined across the block reduction, serpentine rows.
#define ROWS_PER_BLK 16
__global__ __launch_bounds__(256) void sk_pass_kernel(
    const double* __restrict__ Km, const double* __restrict__ v,
    double* __restrict__ vacc, int dir) {
    __shared__ double sred[8];
    __shared__ double ubc;
    int tid = threadIdx.x, lane = tid & 31, wv = tid >> 5;
    // serpentine: dir=1 maps block 0 to the highest rows and walks downward,
    // so each pass starts on the ~192MB L2-resident tail of the previous pass.
    int blk  = dir ? (gridDim.x - 1 - blockIdx.x) : blockIdx.x;
    int base = blk * ROWS_PER_BLK;
    int rowStart = dir ? (base + ROWS_PER_BLK - 1) : base;
    long rowStep = dir ? -(long)K_CODES : (long)K_CODES;
    int c0 = tid * 2;                        // columns 2t,2t+1 (+512 strides)

    double2 vr[4], ca[4];
    #pragma unroll
    for (int c = 0; c < 4; c++) {
        vr[c] = *(const double2*)(v + c0 + 512 * c);
        ca[c] = make_double2(0.0, 0.0);
    }

    const double* rowp = Km + (size_t)rowStart * K_CODES;
    double2 km[4];
    #pragma unroll
    for (int c = 0; c < 4; c++) km[c] = *(const double2*)(rowp + c0 + 512 * c);

    for (int r = 0; r < ROWS_PER_BLK; r++) {
        double2 kn[4];
        if (r + 1 < ROWS_PER_BLK) {          // issue next row before the barrier
            const double* np = rowp + rowStep;
            #pragma unroll
            for (int c = 0; c < 4; c++) kn[c] = *(const double2*)(np + c0 + 512 * c);
        }
        double part = 0.0;
        #pragma unroll
        for (int c = 0; c < 4; c++) part += km[c].x * vr[c].x + km[c].y * vr[c].y;
        #pragma unroll
        for (int off = 16; off; off >>= 1) part += __shfl_down(part, off, 32);
        if (lane == 0) sred[wv] = part;
        __syncthreads();
        if (tid == 0) {
            double s = 0.0;
            #pragma unroll
            for (int q = 0; q < 8; q++) s += sred[q];
            ubc = (1.0 / (double)B_ROWS) / (s + 1e-8);
        }
        __syncthreads();
        double u = ubc;
        #pragma unroll
        for (int c = 0; c < 4; c++) { ca[c].x += u * km[c].x; ca[c].y += u * km[c].y; }
        if (r + 1 < ROWS_PER_BLK) {
            #pragma unroll
            for (int c = 0; c < 4; c++) km[c] = kn[c];
            rowp += rowStep;
        }
    }
    #pragma unroll
    for (int c = 0; c < 4; c++) {
        unsafeAtomicAdd(&vacc[c0 + 512 * c],     ca[c].x);
        unsafeAtomicAdd(&vacc[c0 + 512 * c + 1], ca[c].y);
    }
}

// ---- v update: v = colm/(K^T u + 1e-8); re-zero accumulator for next iter ----
__global__ void sk_colupdate_kernel(double* __restrict__ v, double* __restrict__ vacc) {
    int j = blockIdx.x * 256 + threadIdx.x;
    v[j] = (1.0 / (double)K_CODES) / (vacc[j] + 1e-8);
    vacc[j] = 0.0;
}

// ---- finalize: argmax_j Km[i,j]*v[j]; emb gather; loss = 1.25*mean||x-emb||^2 ----
__global__ __launch_bounds__(256) void finalize_kernel(
    const double* __restrict__ Km, const double* __restrict__ v,
    const float* __restrict__ x, const float* __restrict__ w,
    float* __restrict__ out) {
    __shared__ double smax[8];
    __shared__ int sidx[8];
    __shared__ int sbest;
    int tid = threadIdx.x, lane = tid & 31, wv = tid >> 5;
    int row = blockIdx.x;
    const double* rowp = Km + (size_t)row * K_CODES;
    int c0 = tid * 2;

    double best = -1.0; int bi = 0;       // Km*v > 0 always
    #pragma unroll
    for (int c = 0; c < 4; c++) {
        int j = c0 + 512 * c;
        double2 kmv = *(const double2*)(rowp + j);
        double2 vv  = *(const double2*)(v + j);
        double p0 = kmv.x * vv.x;
        double p1 = kmv.y * vv.y;
        if (p0 > best || (p0 == best && j < bi))     { best = p0; bi = j; }
        if (p1 > best || (p1 == best && j + 1 < bi)) { best = p1; bi = j + 1; }
    }
    #pragma unroll
    for (int off = 16; off; off >>= 1) {
        double ov = __shfl_down(best, off, 32);
        int    oi = __shfl_down(bi,   off, 32);
        if (ov > best || (ov == best && oi < bi)) { best = ov; bi = oi; }
    }
    if (lane == 0) { smax[wv] = best; sidx[wv] = bi; }
    __syncthreads();
    if (tid == 0) {
        double bb = smax[0]; int ii = sidx[0];
        #pragma unroll
        for (int q = 1; q < 8; q++)
            if (smax[q] > bb || (smax[q] == bb && sidx[q] < ii)) { bb = smax[q]; ii = sidx[q]; }
        sbest = ii;
        out[B_ROWS * D_DIM + row] = (float)ii;   // sk_ids as float
    }
    __syncthreads();
    int k = sbest;
    if (tid < 32) {
        float e  = w[(size_t)k * D_DIM + tid];
        float xi = x[(size_t)row * D_DIM + tid];
        out[(size_t)row * D_DIM + tid] = e;      // emb_out == emb in forward
        float df = xi - e;
        float sq = df * df;
        #pragma unroll
        for (int off = 16; off; off >>= 1) sq += __shfl_down(sq, off, 32);
        if (tid == 0)
            unsafeAtomicAdd(&out[LOSS_IDX], 1.25f * sq * (1.0f / (float)B_ROWS));
    }
}

extern "C" void kernel_launch(void* const* d_in, const int* in_sizes, int n_in,
                              void* d_out, int out_size, void* d_ws, size_t ws_size,
                              hipStream_t stream) {
    const float* x = (const float*)d_in[0];
    const float* w = (const float*)d_in[1];
    float* out = (float*)d_out;

    char* ws = (char*)d_ws;
    double* Km  = (double*)ws;                                  // 512 MB
    float*  dist = (float*)(ws + (size_t)B_ROWS * K_CODES * 8); // 268 MB
    char* p = ws + (size_t)B_ROWS * K_CODES * 8 + (size_t)B_ROWS * K_CODES * 4;
    double* v    = (double*)p; p += (size_t)K_CODES * 8;
    double* vacc = (double*)p; p += (size_t)K_CODES * 8;
    float*  xn   = (float*)p;  p += (size_t)B_ROWS * 4;
    float*  cn   = (float*)p;  p += (size_t)K_CODES * 4;
    unsigned* mm = (unsigned*)p;

    init_kernel<<<K_CODES / 256, 256, 0, stream>>>(v, vacc, mm, out);
    norms_kernel<<<B_ROWS / 256, 256, 0, stream>>>(x, w, xn, cn);
    dist_wmma_kernel<<<(B_ROWS / 16) * (K_CODES / 16) / 8, 256, 0, stream>>>(
        x, w, xn, cn, dist, mm);
    km_kernel<<<(int)(((size_t)B_ROWS * K_CODES) / 1024), 256, 0, stream>>>(dist, mm, Km);
    for (int it = 0; it < SK_ITERS_N; ++it) {
        // alternate traversal direction: pass n+1 starts on pass n's L2 tail;
        // it=99 runs reversed, so the forward finalize pass reuses its tail too.
        sk_pass_kernel<<<B_ROWS / ROWS_PER_BLK, 256, 0, stream>>>(Km, v, vacc, it & 1);
        sk_colupdate_kernel<<<K_CODES / 256, 256, 0, stream>>>(v, vacc);
    }
    finalize_kernel<<<B_ROWS, 256, 0, stream>>>(Km, v, x, w, out);
}